// RGCN_49323404427319
// MI455X (gfx1250) — compile-verified
//
#include <hip/hip_runtime.h>
#include <math.h>

// ---------------- problem constants (match reference) ----------------
#define N_NODES 20000
#define IN_F    128
#define NHEAD   2
#define DHEAD   32
#define HDF     64            // NHEAD*DHEAD
#define KHOPS   3
#define RREL    5
#define NEDGE   320000
#define NEG_SLOPE 0.2f

typedef __attribute__((ext_vector_type(2))) float v2f;
typedef __attribute__((ext_vector_type(8))) float v8f;

__device__ __forceinline__ float lrelu(float x) { return x > 0.f ? x : NEG_SLOPE * x; }

// order-preserving float<->uint key for atomicMax on floats
__device__ __forceinline__ unsigned fkey(float f) {
  unsigned u = __float_as_uint(f);
  return (u & 0x80000000u) ? ~u : (u | 0x80000000u);
}
__device__ __forceinline__ float funkey(unsigned u) {
  return __uint_as_float((u & 0x80000000u) ? (u & 0x7fffffffu) : ~u);
}
#define EMAX_INIT 0x007FFFFFu   // fkey(-inf)

__device__ __forceinline__ float waveAllSum(float v) {
  v += __shfl_xor(v, 16, 32);
  v += __shfl_xor(v,  8, 32);
  v += __shfl_xor(v,  4, 32);
  v += __shfl_xor(v,  2, 32);
  v += __shfl_xor(v,  1, 32);
  return v;
}

__device__ __forceinline__ void atomicAddF(float* p, float v) {
  unsafeAtomicAdd(p, v);   // hardware GLOBAL_ATOMIC_ADD_F32
}

// ---------------------------------------------------------------------
// Kernel 1: fused dual GEMM  X0 = h @ fcW , RES = h @ resW
// 256 threads = 8 wave32; block computes 32 rows x 64 cols of both outputs.
// WMMA f32 16x16x4 per ISA VGPR layouts.
// ---------------------------------------------------------------------
__global__ __launch_bounds__(256) void gemm_proj_kernel(
    const float* __restrict__ h, const float* __restrict__ fcw,
    const float* __restrict__ resw, float* __restrict__ X0,
    float* __restrict__ RES) {
  __shared__ float As[32][132];   // 132*4B row stride stays 16B aligned
  const int t = threadIdx.x;
  const int rowBase = blockIdx.x * 32;   // N=20000 divides by 32 exactly (625 blocks)

  // coalesced float4 load of the 32x128 h-tile into LDS
  const float4* hp = (const float4*)(h + rowBase * IN_F);
  for (int i = 0; i < 4; ++i) {
    int idx = t + i * 256;        // 0..1023 float4 slots
    int row = idx >> 5;
    int c4  = idx & 31;
    float4 v = hp[row * 32 + c4];
    *(float4*)&As[row][c4 * 4] = v;
  }
  __syncthreads();

  const int wid  = t >> 5;
  const int lane = t & 31;
  const int rsub = wid >> 2;          // 0/1 -> which 16-row half
  const int ct   = wid & 3;           // 16-col tile of 64
  const int m    = lane & 15;
  const int khalf = (lane >> 4) << 1; // 0 or 2 (A/B K split per ISA layout)
  const int col  = ct * 16 + m;

  v8f accF = {};
  v8f accR = {};
  for (int k = 0; k < IN_F; k += 4) {
    v2f a;
    a.x = As[rsub * 16 + m][k + khalf];
    a.y = As[rsub * 16 + m][k + khalf + 1];
    v2f bf;
    bf.x = fcw[(k + khalf) * HDF + col];
    bf.y = fcw[(k + khalf + 1) * HDF + col];
    v2f br;
    br.x = resw[(k + khalf) * HDF + col];
    br.y = resw[(k + khalf + 1) * HDF + col];
    accF = __builtin_amdgcn_wmma_f32_16x16x4_f32(false, a, false, bf,
                                                 (short)0, accF, false, false);
    accR = __builtin_amdgcn_wmma_f32_16x16x4_f32(false, a, false, br,
                                                 (short)0, accR, false, false);
  }

  // D layout: VGPR j -> M=j (lanes 0-15) / M=8+j (lanes 16-31); N = lane&15
  const int rlo  = rowBase + rsub * 16 + ((lane >> 4) ? 8 : 0);
  const int ccol = ct * 16 + (lane & 15);
  for (int j = 0; j < 8; ++j) {
    int row = rlo + j;
    X0[row * HDF + ccol]  = accF[j];
    RES[row * HDF + ccol] = accR[j];
  }
}

// ---------------------------------------------------------------------
// Kernel 2: attention logits  el[n,h] = <x[n,h,:], attn_l[h,:]>  (and er)
// ---------------------------------------------------------------------
__global__ void attn_logit_kernel(const float* __restrict__ X0,
                                  const float* __restrict__ al,
                                  const float* __restrict__ ar,
                                  float* __restrict__ EL,
                                  float* __restrict__ ER) {
  int gid = blockIdx.x * blockDim.x + threadIdx.x;
  if (gid >= N_NODES * NHEAD) return;
  int n = gid >> 1, hh = gid & 1;
  const float* x  = X0 + n * HDF + hh * DHEAD;
  const float* wl = al + hh * DHEAD;
  const float* wr = ar + hh * DHEAD;
  float sl = 0.f, sr = 0.f;
#pragma unroll
  for (int d = 0; d < DHEAD; ++d) {
    float v = x[d];
    sl += v * wl[d];
    sr += v * wr[d];
  }
  EL[gid] = sl;
  ER[gid] = sr;
}

// ---------------------------------------------------------------------
// Kernel 3: per-relation scratch init (zero hops 1..3, EMAX keys, DENOM)
// ---------------------------------------------------------------------
__global__ void init_rel_kernel(float* __restrict__ hops123,
                                unsigned* __restrict__ EMAX,
                                float* __restrict__ DENOM, int nzero) {
  int gid = blockIdx.x * blockDim.x + threadIdx.x;
  if (gid < nzero) hops123[gid] = 0.f;
  if (gid < N_NODES * NHEAD) {
    EMAX[gid]  = EMAX_INIT;
    DENOM[gid] = 0.f;
  }
}

// ---------------------------------------------------------------------
// Kernels 4-6: segment softmax over incoming edges of each dst node
// ---------------------------------------------------------------------
__global__ void edge_logit_kernel(const int* __restrict__ src,
                                  const int* __restrict__ dst,
                                  const float* __restrict__ EL,
                                  const float* __restrict__ ER,
                                  float* __restrict__ ALPHA,
                                  unsigned* __restrict__ EMAX) {
  int gid = blockIdx.x * blockDim.x + threadIdx.x;
  if (gid >= NEDGE * NHEAD) return;
  int eid = gid >> 1, hh = gid & 1;
  int s = src[eid], d = dst[eid];
  float e = lrelu(EL[s * NHEAD + hh] + ER[d * NHEAD + hh]);
  ALPHA[gid] = e;
  atomicMax(&EMAX[d * NHEAD + hh], fkey(e));
}

__global__ void edge_exp_kernel(const int* __restrict__ dst,
                                float* __restrict__ ALPHA,
                                const unsigned* __restrict__ EMAX,
                                float* __restrict__ DENOM) {
  int gid = blockIdx.x * blockDim.x + threadIdx.x;
  if (gid >= NEDGE * NHEAD) return;
  int eid = gid >> 1, hh = gid & 1;
  int d = dst[eid];
  float m  = funkey(EMAX[d * NHEAD + hh]);
  float ee = __expf(ALPHA[gid] - m);
  ALPHA[gid] = ee;
  atomicAddF(&DENOM[d * NHEAD + hh], ee);
}

__global__ void edge_norm_kernel(const int* __restrict__ dst,
                                 float* __restrict__ ALPHA,
                                 const float* __restrict__ DENOM) {
  int gid = blockIdx.x * blockDim.x + threadIdx.x;
  if (gid >= NEDGE * NHEAD) return;
  int eid = gid >> 1, hh = gid & 1;
  int d = dst[eid];
  ALPHA[gid] = ALPHA[gid] / fmaxf(DENOM[d * NHEAD + hh], 1e-9f);
}

// ---------------------------------------------------------------------
// Kernel 7: one diffusion hop  Xout[dst] += alpha * Xin[src]
// 64 consecutive threads own one edge's 64 features -> gather/scatter is
// one 128B segment per wave; fp32 hardware atomics for the scatter.
// ---------------------------------------------------------------------
__global__ void spmm_hop_kernel(const int* __restrict__ src,
                                const int* __restrict__ dst,
                                const float* __restrict__ ALPHA,
                                const float* __restrict__ Xin,
                                float* __restrict__ Xout) {
  int gid = blockIdx.x * blockDim.x + threadIdx.x;
  if (gid >= NEDGE * HDF) return;
  int eid  = gid >> 6;
  int comp = gid & 63;
  int hh   = comp >> 5;
  int s = src[eid], d = dst[eid];
  float v = ALPHA[eid * NHEAD + hh] * Xin[s * HDF + comp];
  atomicAddF(&Xout[d * HDF + comp], v);
}

// ---------------------------------------------------------------------
// Kernel 8: hop-norm + hop attention + residual + relation/head combine.
// One wave32 per node; lane = feature d. Accumulates into d_out.
// ---------------------------------------------------------------------
__global__ __launch_bounds__(256) void combine_kernel(
    const float* __restrict__ X,      // hops [4][N,H,D] flat, stride NHD
    const float* __restrict__ RES,    // [N,H,D]
    const float* __restrict__ hal,    // [H,D] (relation-offset)
    const float* __restrict__ har,    // [H,D]
    const float* __restrict__ w_rel,  // [R,R]
    const float* __restrict__ b_rel,  // [R]
    float* __restrict__ out, int r, int first) {
  const int NHD = N_NODES * HDF;
  int wgid = (blockIdx.x * blockDim.x + threadIdx.x) >> 5;
  int lane = threadIdx.x & 31;
  if (wgid >= N_NODES) return;
  int n = wgid;

  float wrel = b_rel[r];
#pragma unroll
  for (int j = 0; j < RREL; ++j) wrel += w_rel[r * RREL + j];

  float acc = 0.f;
#pragma unroll
  for (int hh = 0; hh < NHEAD; ++hh) {
    int base = n * HDF + hh * DHEAD + lane;
    float v0 = X[0 * NHD + base];
    float v1 = X[1 * NHD + base];
    float v2 = X[2 * NHD + base];
    float v3 = X[3 * NHD + base];
    // L2-normalize each hop over D (wave reduction across 32 lanes)
    v0 *= 1.f / fmaxf(sqrtf(waveAllSum(v0 * v0)), 1e-9f);
    v1 *= 1.f / fmaxf(sqrtf(waveAllSum(v1 * v1)), 1e-9f);
    v2 *= 1.f / fmaxf(sqrtf(waveAllSum(v2 * v2)), 1e-9f);
    v3 *= 1.f / fmaxf(sqrtf(waveAllSum(v3 * v3)), 1e-9f);

    float aL = hal[hh * DHEAD + lane];
    float aR = har[hh * DHEAD + lane];
    float hr = waveAllSum(v0 * aR);
    float l0 = lrelu(waveAllSum(v0 * aL) + hr);
    float l1 = lrelu(waveAllSum(v1 * aL) + hr);
    float l2 = lrelu(waveAllSum(v2 * aL) + hr);
    float l3 = lrelu(waveAllSum(v3 * aL) + hr);

    float mx = fmaxf(fmaxf(l0, l1), fmaxf(l2, l3));
    float w0 = __expf(l0 - mx), w1 = __expf(l1 - mx);
    float w2 = __expf(l2 - mx), w3 = __expf(l3 - mx);
    float inv = 1.f / (w0 + w1 + w2 + w3);

    float o = (w0 * v0 + w1 * v1 + w2 * v2 + w3 * v3) * inv + RES[base];
    acc += 0.5f * o;   // mean over NHEAD=2
  }

  int oidx = n * DHEAD + lane;
  float val = wrel * acc;
  if (first) out[oidx] = val;
  else       out[oidx] += val;
}

// ---------------------------------------------------------------------
extern "C" void kernel_launch(void* const* d_in, const int* in_sizes, int n_in,
                              void* d_out, int out_size, void* d_ws, size_t ws_size,
                              hipStream_t stream) {
  (void)in_sizes; (void)n_in; (void)out_size; (void)ws_size;

  const float* h       = (const float*)d_in[0];
  const int*   src_idx = (const int*)d_in[1];     // [R,E]
  const int*   dst_idx = (const int*)d_in[2];     // [R,E]
  const float* fc_w    = (const float*)d_in[3];   // [R,IN,HD]
  const float* res_w   = (const float*)d_in[4];   // [R,IN,HD]
  const float* attn_l  = (const float*)d_in[5];   // [R,H,D]
  const float* attn_r  = (const float*)d_in[6];
  const float* hop_al  = (const float*)d_in[7];
  const float* hop_ar  = (const float*)d_in[8];
  const float* w_rel   = (const float*)d_in[9];   // [R,R]
  const float* b_rel   = (const float*)d_in[10];  // [R]
  float* out = (float*)d_out;                     // [N,D]

  const int NHD = N_NODES * HDF;      // 1,280,000
  const int NH  = N_NODES * NHEAD;    // 40,000

  float*    W     = (float*)d_ws;
  float*    X     = W;                 // hops 0..3 : 4*NHD
  float*    RESb  = W + 4 * NHD;       // NHD
  float*    EL    = W + 5 * NHD;       // NH
  float*    ER    = EL + NH;           // NH
  unsigned* EMAX  = (unsigned*)(ER + NH);          // NH
  float*    DENOM = (float*)(EMAX + NH);           // NH
  float*    ALPHA = DENOM + NH;        // E*H = 640,000
  // total: 7.2M floats = 28.8 MB of d_ws

  const dim3 B256(256);
  const int gGemm  = N_NODES / 32;                       // 625 (exact)
  const int gAttn  = (NH + 255) / 256;
  const int gInit  = (KHOPS * NHD + 255) / 256;
  const int gEdge  = (NEDGE * NHEAD + 255) / 256;        // 2500
  const int gSpmm  = (NEDGE * HDF + 255) / 256;          // 80000
  const int gComb  = (N_NODES * 32 + 255) / 256;         // 2500 (8 waves/blk)

  for (int r = 0; r < RREL; ++r) {
    const float* fcw  = fc_w  + (size_t)r * IN_F * HDF;
    const float* rsw  = res_w + (size_t)r * IN_F * HDF;
    const float* al   = attn_l + (size_t)r * NHEAD * DHEAD;
    const float* ar   = attn_r + (size_t)r * NHEAD * DHEAD;
    const float* hal  = hop_al + (size_t)r * NHEAD * DHEAD;
    const float* har  = hop_ar + (size_t)r * NHEAD * DHEAD;
    const int*   src  = src_idx + (size_t)r * NEDGE;
    const int*   dst  = dst_idx + (size_t)r * NEDGE;

    gemm_proj_kernel<<<gGemm, B256, 0, stream>>>(h, fcw, rsw, X, RESb);
    attn_logit_kernel<<<gAttn, B256, 0, stream>>>(X, al, ar, EL, ER);
    init_rel_kernel<<<gInit, B256, 0, stream>>>(X + NHD, EMAX, DENOM, KHOPS * NHD);
    edge_logit_kernel<<<gEdge, B256, 0, stream>>>(src, dst, EL, ER, ALPHA, EMAX);
    edge_exp_kernel<<<gEdge, B256, 0, stream>>>(dst, ALPHA, EMAX, DENOM);
    edge_norm_kernel<<<gEdge, B256, 0, stream>>>(dst, ALPHA, DENOM);
    for (int k = 1; k <= KHOPS; ++k) {
      spmm_hop_kernel<<<gSpmm, B256, 0, stream>>>(src, dst, ALPHA,
                                                  X + (size_t)(k - 1) * NHD,
                                                  X + (size_t)k * NHD);
    }
    combine_kernel<<<gComb, B256, 0, stream>>>(X, RESb, hal, har, w_rel, b_rel,
                                               out, r, r == 0 ? 1 : 0);
  }
}